// Rtfm_89730456748399
// MI455X (gfx1250) — compile-verified
//
#include <hip/hip_runtime.h>
#include <math.h>

#define B_ 32
#define T_ 256
#define F_ 2048
#define K_ 16
#define CH 64

typedef __attribute__((ext_vector_type(2))) float v2f;
typedef __attribute__((ext_vector_type(4))) float v4f;
typedef __attribute__((ext_vector_type(8))) float v8f;

// ---------------------------------------------------------------------------
// Kernel 1: fused drop = fmagn*mask, top-16 per row (lowest-index tie-break,
// matching jax.lax.top_k), and vls = mean of gathered sls.
// blocks 0..31 -> abnormal rows, 32..63 -> normal rows.
// ws_idx rows: [0..31]=abn, [32..63]=norm. ws_vls: [0..31]=norm, [32..63]=abn
// (reference concat order).
// ---------------------------------------------------------------------------
__global__ void topk_vls_kernel(const float* __restrict__ fmagn_a,
                                const float* __restrict__ mask_a,
                                const float* __restrict__ sls_a,
                                const float* __restrict__ fmagn_n,
                                const float* __restrict__ mask_n,
                                const float* __restrict__ sls_n,
                                int* __restrict__ ws_idx,
                                float* __restrict__ ws_vls) {
    const int bb = blockIdx.x;          // 0..63
    const int t  = threadIdx.x;         // 0..255
    const bool is_abn = (bb < B_);
    const int b = is_abn ? bb : (bb - B_);
    const float* fm = is_abn ? fmagn_a : fmagn_n;
    const float* mk = is_abn ? mask_a  : mask_n;
    const float* sl = is_abn ? sls_a   : sls_n;

    __shared__ float sv[T_];
    __shared__ float rv[T_];
    __shared__ int   ri[T_];
    __shared__ int   sel[K_];

    sv[t] = fm[b * T_ + t] * mk[b * T_ + t];
    __syncthreads();

    for (int k = 0; k < K_; ++k) {
        rv[t] = sv[t];
        ri[t] = t;
        __syncthreads();
        for (int s = T_ / 2; s > 0; s >>= 1) {
            if (t < s) {
                const float ov = rv[t + s];
                const int   oi = ri[t + s];
                if (ov > rv[t] || (ov == rv[t] && oi < ri[t])) {
                    rv[t] = ov;
                    ri[t] = oi;
                }
            }
            __syncthreads();
        }
        if (t == 0) {
            sel[k] = ri[0];
            sv[ri[0]] = -3.0e38f;       // remove from further selection
        }
        __syncthreads();
    }

    if (t < K_) ws_idx[bb * K_ + t] = sel[t];
    if (t == 0) {
        float s = 0.f;
        for (int k = 0; k < K_; ++k) s += sl[b * T_ + sel[k]];
        const float vls = s * (1.f / K_);
        ws_vls[is_abn ? (B_ + b) : b] = vls;   // concat([norm, abn]) order
    }
}

// ---------------------------------------------------------------------------
// Kernel 2: per group of 16 batch rows, gather selected rows (float4 /
// global_load_b128 — contiguous, 16B-aligned), mean over K into a skewed LDS
// chunk, then compute all 16 L2 norms at once via a chained f32 WMMA Gram:
//   C(16x16) += Achunk(16x4) * Achunk^T(4x16),  diag(C)[m] = ||afeat_m||^2.
// For f32 16x16x4, the documented A layout (lanes 0-15: M=lane, v0/v1=K0/K1;
// lanes 16-31: K2/K3) and B layout (lane halves = K halves, lane%16 = N) make
// the A and B register images identical when B = A^T of the same data, so the
// same v2f feeds both operands.
// grid: 4 blocks x 32 threads (one wave; EXEC all ones around WMMA).
//   g=0: abn b 0..15, g=1: abn b 16..31, g=2: norm b 0..15, g=3: norm b 16..31
// ---------------------------------------------------------------------------
__global__ void mean_gram_kernel(const float* __restrict__ feats_a,  // feats[-1]
                                 const float* __restrict__ feats_n,  // feats[-1]
                                 const int*   __restrict__ ws_idx,
                                 float*       __restrict__ ws_norms) {
    const int g    = blockIdx.x;
    const int lane = threadIdx.x;       // 0..31
    const bool is_abn = (g < 2);
    const int b_base = (g & 1) * 16;
    const float* feats = is_abn ? feats_a : feats_n;
    const int idx_row_base = (is_abn ? 0 : B_) + b_base;

    __shared__ int   sidx[16][K_];
    // +4 skew: row byte stride 272 = 17*16 -> rows on distinct banks, and
    // every chunk[m][4*i] remains 16B-aligned for v4f stores.
    __shared__ float chunk[16][CH + 4];

    for (int e = lane; e < 16 * K_; e += 32) {
        sidx[e >> 4][e & 15] = ws_idx[(idx_row_base + (e >> 4)) * K_ + (e & 15)];
    }
    __syncthreads();

    v8f c = {0.f, 0.f, 0.f, 0.f, 0.f, 0.f, 0.f, 0.f};
    const int r  = lane & 15;                 // M for A / N for B
    const int kb = (lane < 16) ? 0 : 2;       // K half selected by lane group

    const int NG = CH / 4;                    // float4 groups per row (16)

    for (int f0 = 0; f0 < F_; f0 += CH) {
        // gather + mean: 16 rows x CH features per chunk, 16B vector loads.
        for (int e = lane; e < 16 * NG; e += 32) {
            const int m  = e >> 4;            // row within group
            const int f4 = (e & (NG - 1)) * 4;
            const int b  = b_base + m;
            v4f s = {0.f, 0.f, 0.f, 0.f};
#pragma unroll
            for (int k = 0; k < K_; ++k) {
                const v4f v = *(const v4f*)&feats[((size_t)b * T_ + sidx[m][k]) * F_ +
                                                  (f0 + f4)];
                s += v;
            }
            *(v4f*)&chunk[m][f4] = s * (1.f / K_);
        }
        __syncthreads();

        if (f0 + CH < F_) {                   // gfx1250 global_prefetch_b8
            __builtin_prefetch(&feats[((size_t)b_base * T_ + sidx[0][0]) * F_ +
                                      (f0 + CH)], 0, 1);
        }

        // chained f32 WMMA: K=4 per op, 16 ops per chunk
#pragma unroll
        for (int fo = 0; fo < CH; fo += 4) {
            v2f a;
            a.x = chunk[r][fo + kb];
            a.y = chunk[r][fo + kb + 1];
            c = __builtin_amdgcn_wmma_f32_16x16x4_f32(
                    /*neg_a=*/false, a, /*neg_b=*/false, a,
                    /*c_mod=*/(short)0, c, /*reuse_a=*/false, /*reuse_b=*/false);
        }
        __syncthreads();
    }

    // diagonal of the Gram matrix -> L2 norms.
    // C/D layout: lane<16: M=vgpr_r, N=lane; lane>=16: M=vgpr_r+8, N=lane-16.
#pragma unroll
    for (int rr = 0; rr < 8; ++rr) {
        const int M = rr + ((lane >= 16) ? 8 : 0);
        const int N = lane & 15;
        if (M == N) {
            ws_norms[(is_abn ? 0 : B_) + b_base + M] = sqrtf(c[rr]);
        }
    }
}

// ---------------------------------------------------------------------------
// Kernel 3: final scalar losses. ws_norms: [0..31]=abn, [32..63]=norm.
// ws_vls already in concat([norm, abn]) order matching label.
// ---------------------------------------------------------------------------
__global__ void final_kernel(const float* __restrict__ ws_vls,
                             const float* __restrict__ ws_norms,
                             const float* __restrict__ label,
                             float* __restrict__ out) {
    const int t = threadIdx.x;          // 0..63
    __shared__ float red[64];

    // BCE with torch-style clamp(log, -100)
    const float v   = ws_vls[t];
    const float lab = label[t];
    const float lp  = fmaxf(logf(v), -100.f);
    const float l1  = fmaxf(log1pf(-v), -100.f);
    red[t] = lab * lp + (1.f - lab) * l1;
    __syncthreads();
    for (int s = 32; s > 0; s >>= 1) {
        if (t < s) red[t] += red[t + s];
        __syncthreads();
    }
    const float bce = -red[0] * (1.f / 64.f);
    __syncthreads();

    // RTFM: duplicated halves identical -> mean over 2B == mean over B
    float rt = 0.f;
    if (t < B_) {
        const float la = fabsf(100.f - ws_norms[t]);        // abn
        const float ln = ws_norms[B_ + t];                  // norm
        const float s  = la + ln;
        rt = s * s;
    }
    red[t] = rt;
    __syncthreads();
    for (int s = 32; s > 0; s >>= 1) {
        if (t < s) red[t] += red[t + s];
        __syncthreads();
    }
    if (t == 0) {
        out[0] = 1e-4f * (red[0] * (1.f / B_));
        out[1] = bce;
    }
}

extern "C" void kernel_launch(void* const* d_in, const int* in_sizes, int n_in,
                              void* d_out, int out_size, void* d_ws, size_t ws_size,
                              hipStream_t stream) {
    const float* abnr_fmagn = (const float*)d_in[0];
    const float* norm_fmagn = (const float*)d_in[1];
    const float* abnr_feats = (const float*)d_in[2];
    const float* norm_feats = (const float*)d_in[3];
    const float* abnr_sls   = (const float*)d_in[4];
    const float* norm_sls   = (const float*)d_in[5];
    const float* label      = (const float*)d_in[6];
    const float* mask_abn   = (const float*)d_in[7];
    const float* mask_norm  = (const float*)d_in[8];
    float* out = (float*)d_out;

    int*   ws_idx   = (int*)d_ws;                                    // 64*16 ints
    float* ws_vls   = (float*)((char*)d_ws + 64 * K_ * sizeof(int)); // 64 f32
    float* ws_norms = ws_vls + 64;                                   // 64 f32

    const size_t last = (size_t)(10 - 1) * B_ * T_ * F_;             // feats[-1]

    topk_vls_kernel<<<64, 256, 0, stream>>>(abnr_fmagn, mask_abn, abnr_sls,
                                            norm_fmagn, mask_norm, norm_sls,
                                            ws_idx, ws_vls);
    mean_gram_kernel<<<4, 32, 0, stream>>>(abnr_feats + last, norm_feats + last,
                                           ws_idx, ws_norms);
    final_kernel<<<1, 64, 0, stream>>>(ws_vls, ws_norms, label, out);
}